// InstanceAwareContrast_51256139710649
// MI455X (gfx1250) — compile-verified
//
#include <hip/hip_runtime.h>
#include <hip/hip_bf16.h>

// InstanceAwareContrast on MI455X (gfx1250, wave32).
// Memory-bound: 128 MB fp32 traversed twice; second pass hits 192MB L2.
// Scoring pass uses native fp32 WMMA (v_wmma_f32_16x16x4_f32) => exact fp32 GEMM.

#define TAU       0.07f
#define LAMBDA_CF 0.5f
#define K1        17     // segments 0..16
#define KPAD      32     // padded centroid rows for WMMA N-tiles
#define DDIM      256

typedef float v2f __attribute__((ext_vector_type(2)));
typedef float v8f __attribute__((ext_vector_type(8)));

// ---------------- workspace layout (floats) ----------------
// [0)       g_cnt[32]
// [32)      g_sum_dp[32*256]   (only first 17 rows used)
// [8224)    g_sum_cf[32*256]
// [16416)   g_loss_t[32]
// [16448)   g_loss_cf[32]
// [16480)   mu_dp[32*256]
// [24672)   mu_cf[32*256]
// [32864)   rinv_dp[N]
// [32864+N) rinv_cf[N]
#define OFF_CNT     0
#define OFF_SUM_DP  32
#define OFF_SUM_CF  (32 + 32*256)
#define OFF_LOSS_T  (32 + 2*32*256)
#define OFF_LOSS_CF (OFF_LOSS_T + 32)
#define OFF_MU_DP   (OFF_LOSS_CF + 32)
#define OFF_MU_CF   (OFF_MU_DP + 32*256)
#define OFF_RINV_DP (OFF_MU_CF + 32*256)

__device__ __forceinline__ float softplus_stable(float z) {
    // log(1+exp(z)), stable both directions
    return (z > 0.0f) ? (z + log1pf(__expf(-z))) : log1pf(__expf(z));
}

// ---------------- kernel 0: zero accumulators ----------------
__global__ void k_zero(float* ws, int n) {
    int i = blockIdx.x * blockDim.x + threadIdx.x;
    if (i < n) ws[i] = 0.0f;
}

// ---------------- kernel 1: normalize rows + segment sums ----------------
// One wave per row (grid-stride). Wave32 shuffle reduction for the norm,
// LDS ds_add_f32 scatter into per-block [17][256] accumulators, then a
// single global_atomic_add_f32 flush per block.
__global__ void k_accum(const float* __restrict__ dp,
                        const float* __restrict__ cf,
                        const int*   __restrict__ labels,
                        float* __restrict__ ws, int nrows) {
    __shared__ float s_dp[K1 * DDIM];
    __shared__ float s_cf[K1 * DDIM];
    __shared__ float s_cnt[32];

    for (int i = threadIdx.x; i < K1 * DDIM; i += blockDim.x) { s_dp[i] = 0.0f; s_cf[i] = 0.0f; }
    if (threadIdx.x < 32) s_cnt[threadIdx.x] = 0.0f;
    __syncthreads();

    const int lane  = threadIdx.x & 31;
    const int wid   = threadIdx.x >> 5;
    const int wpb   = blockDim.x >> 5;
    const int gwid  = blockIdx.x * wpb + wid;
    const int nwave = gridDim.x * wpb;

    float* rinv_dp = ws + OFF_RINV_DP;
    float* rinv_cf = ws + OFF_RINV_DP + nrows;

    for (int row = gwid; row < nrows; row += nwave) {
        const int lab = labels[row];

        // ---- dp row ----
        {
            const float* p = dp + (size_t)row * DDIM;
            float v[8]; float ss = 0.0f;
            #pragma unroll
            for (int j = 0; j < 8; ++j) { v[j] = p[lane + 32 * j]; ss += v[j] * v[j]; }
            #pragma unroll
            for (int off = 16; off > 0; off >>= 1) ss += __shfl_xor(ss, off, 32);
            const float sc = 1.0f / fmaxf(sqrtf(ss), 1e-12f);
            if (lane == 0) rinv_dp[row] = sc;
            #pragma unroll
            for (int j = 0; j < 8; ++j) atomicAdd(&s_dp[lab * DDIM + lane + 32 * j], v[j] * sc);
        }
        // ---- cf row ----
        {
            const float* p = cf + (size_t)row * DDIM;
            float v[8]; float ss = 0.0f;
            #pragma unroll
            for (int j = 0; j < 8; ++j) { v[j] = p[lane + 32 * j]; ss += v[j] * v[j]; }
            #pragma unroll
            for (int off = 16; off > 0; off >>= 1) ss += __shfl_xor(ss, off, 32);
            const float sc = 1.0f / fmaxf(sqrtf(ss), 1e-12f);
            if (lane == 0) rinv_cf[row] = sc;
            #pragma unroll
            for (int j = 0; j < 8; ++j) atomicAdd(&s_cf[lab * DDIM + lane + 32 * j], v[j] * sc);
        }
        if (lane == 0) atomicAdd(&s_cnt[lab], 1.0f);
    }
    __syncthreads();

    for (int i = threadIdx.x; i < K1 * DDIM; i += blockDim.x) {
        unsafeAtomicAdd(ws + OFF_SUM_DP + i, s_dp[i]);
        unsafeAtomicAdd(ws + OFF_SUM_CF + i, s_cf[i]);
    }
    if (threadIdx.x < 32) unsafeAtomicAdd(ws + OFF_CNT + threadIdx.x, s_cnt[threadIdx.x]);
}

// ---------------- kernel 2: centroids mu = l2norm(sum/cnt), padded to 32 rows ----
__global__ void k_mu(float* __restrict__ ws) {
    __shared__ float red[DDIM];
    const int d = threadIdx.x;  // blockDim.x == 256 == DDIM
    const float* g_cnt = ws + OFF_CNT;
    for (int s = 0; s < KPAD; ++s) {
        float vdp = 0.0f, vcf = 0.0f;
        if (s < K1) {
            const float c = fmaxf(g_cnt[s], 1.0f);
            vdp = ws[OFF_SUM_DP + s * DDIM + d] / c;
            vcf = ws[OFF_SUM_CF + s * DDIM + d] / c;
        }
        // reduce |vdp|^2
        red[d] = vdp * vdp; __syncthreads();
        for (int off = 128; off > 0; off >>= 1) { if (d < off) red[d] += red[d + off]; __syncthreads(); }
        const float ndp = fmaxf(sqrtf(red[0]), 1e-12f); __syncthreads();
        // reduce |vcf|^2
        red[d] = vcf * vcf; __syncthreads();
        for (int off = 128; off > 0; off >>= 1) { if (d < off) red[d] += red[d + off]; __syncthreads(); }
        const float ncf = fmaxf(sqrtf(red[0]), 1e-12f); __syncthreads();

        ws[OFF_MU_DP + s * DDIM + d] = vdp / ndp;
        ws[OFF_MU_CF + s * DDIM + d] = vcf / ncf;
    }
}

// ---------------- kernel 3: WMMA scoring pass ----------------
// Each wave owns a 16-row tile:  S[16][32] = tile(16x256) @ Mu^T(256x32)
// via v_wmma_f32_16x16x4_f32, K in chunks of 4 (64 chunks x 2 N-tiles).
// A frag layout: lane holds A[lane&15][2*(lane>>4)+j] in vgpr j (float2 load).
// B frag layout: lane holds B[2*(lane>>4)+j][lane&15] = Mu[lane&15][k] (float2).
// C/D layout: vgpr j = D[j + 8*(lane>>4)][lane&15].
__global__ void k_score(const float* __restrict__ dp,
                        const float* __restrict__ cf,
                        const int*   __restrict__ labels,
                        float* __restrict__ ws, int nrows) {
    __shared__ float Sdp[8][16][KPAD];
    __shared__ float Scf[8][16][KPAD];
    __shared__ float lacc_t[32];
    __shared__ float lacc_cf[32];

    if (threadIdx.x < 32) { lacc_t[threadIdx.x] = 0.0f; lacc_cf[threadIdx.x] = 0.0f; }
    __syncthreads();

    const int lane = threadIdx.x & 31;
    const int w    = threadIdx.x >> 5;
    const int n    = lane & 15;
    const int h    = lane >> 4;
    const int tile = blockIdx.x * (blockDim.x >> 5) + w;
    const int ntiles = nrows >> 4;

    const float* mu_dp = ws + OFF_MU_DP;
    const float* mu_cf = ws + OFF_MU_CF;
    const float* rinv_dp = ws + OFF_RINV_DP;
    const float* rinv_cf = ws + OFF_RINV_DP + nrows;

    if (tile < ntiles) {
        const int r0 = tile * 16;
        const size_t arow = (size_t)(r0 + n) * DDIM;  // A-matrix row for this lane
        v8f acc_dp0 = {}, acc_dp1 = {}, acc_cf0 = {}, acc_cf1 = {};

        #pragma unroll 4
        for (int k = 0; k < DDIM; k += 4) {
            const int kk = k + 2 * h;
            const v2f a_dp = *(const v2f*)(dp + arow + kk);
            const v2f a_cf = *(const v2f*)(cf + arow + kk);
            const v2f b_dp0 = *(const v2f*)(mu_dp + (size_t)n * DDIM + kk);
            const v2f b_dp1 = *(const v2f*)(mu_dp + (size_t)(n + 16) * DDIM + kk);
            const v2f b_cf0 = *(const v2f*)(mu_cf + (size_t)n * DDIM + kk);
            const v2f b_cf1 = *(const v2f*)(mu_cf + (size_t)(n + 16) * DDIM + kk);
            acc_dp0 = __builtin_amdgcn_wmma_f32_16x16x4_f32(false, a_dp, false, b_dp0, (short)0, acc_dp0, false, false);
            acc_dp1 = __builtin_amdgcn_wmma_f32_16x16x4_f32(false, a_dp, false, b_dp1, (short)0, acc_dp1, false, false);
            acc_cf0 = __builtin_amdgcn_wmma_f32_16x16x4_f32(false, a_cf, false, b_cf0, (short)0, acc_cf0, false, false);
            acc_cf1 = __builtin_amdgcn_wmma_f32_16x16x4_f32(false, a_cf, false, b_cf1, (short)0, acc_cf1, false, false);
        }

        #pragma unroll
        for (int j = 0; j < 8; ++j) {
            Sdp[w][j + 8 * h][n]      = acc_dp0[j];
            Sdp[w][j + 8 * h][16 + n] = acc_dp1[j];
            Scf[w][j + 8 * h][n]      = acc_cf0[j];
            Scf[w][j + 8 * h][16 + n] = acc_cf1[j];
        }
    }
    __syncthreads();

    if (tile < ntiles && lane < 16) {
        const int row = tile * 16 + lane;
        const int lab = labels[row];
        const float sdp = rinv_dp[row];
        const float scf = rinv_cf[row];
        // loss_t: softplus((dpn.mu_bg - dpn.mu_lab)/tau)
        const float pos_t = Sdp[w][lane][lab] * sdp;
        const float neg_t = Sdp[w][lane][0]   * sdp;
        const float lt = softplus_stable((neg_t - pos_t) / TAU);
        // loss_cf: softplus((cfn.mu_lab - cfn.mu_bg)/tau)
        const float oth_c = Scf[w][lane][lab] * scf;
        const float pos_c = Scf[w][lane][0]   * scf;
        const float lc = softplus_stable((oth_c - pos_c) / TAU);
        atomicAdd(&lacc_t[lab], lt);
        atomicAdd(&lacc_cf[lab], lc);
    }
    __syncthreads();

    if (threadIdx.x < 32) {
        unsafeAtomicAdd(ws + OFF_LOSS_T  + threadIdx.x, lacc_t[threadIdx.x]);
        unsafeAtomicAdd(ws + OFF_LOSS_CF + threadIdx.x, lacc_cf[threadIdx.x]);
    }
}

// ---------------- kernel 4: final scalar ----------------
__global__ void k_final(const float* __restrict__ ws, float* __restrict__ out) {
    if (threadIdx.x == 0 && blockIdx.x == 0) {
        const float* g_cnt = ws + OFF_CNT;
        float nv = 0.0f, st = 0.0f, sc = 0.0f;
        for (int s = 1; s < K1; ++s) {
            const float c = g_cnt[s];
            if (c >= 3.0f) {
                nv += 1.0f;
                const float safe = fmaxf(c, 1.0f);
                st += ws[OFF_LOSS_T  + s] / safe;
                sc += ws[OFF_LOSS_CF + s] / safe;
            }
        }
        nv = fmaxf(nv, 1.0f);
        out[0] = st / nv + LAMBDA_CF * (sc / nv);
    }
}

extern "C" void kernel_launch(void* const* d_in, const int* in_sizes, int n_in,
                              void* d_out, int out_size, void* d_ws, size_t ws_size,
                              hipStream_t stream) {
    const float* dp     = (const float*)d_in[0];
    const float* cf     = (const float*)d_in[1];
    const int*   labels = (const int*)d_in[2];
    float* ws  = (float*)d_ws;
    float* out = (float*)d_out;
    const int nrows = in_sizes[2];  // N = 65536

    // 0) zero accumulators (cnt + both sums + loss accs)
    const int nzero = OFF_LOSS_CF + 32;  // everything up through loss arrays
    k_zero<<<(nzero + 255) / 256, 256, 0, stream>>>(ws, nzero);

    // 1) per-row normalize + segment sums (one wave per row, grid-stride)
    k_accum<<<512, 256, 0, stream>>>(dp, cf, labels, ws, nrows);

    // 2) centroids (single block, 256 threads == D)
    k_mu<<<1, 256, 0, stream>>>(ws);

    // 3) WMMA scoring: 16-row tile per wave, 8 waves per block
    const int ntiles = nrows >> 4;
    k_score<<<(ntiles + 7) / 8, 256, 0, stream>>>(dp, cf, labels, ws, nrows);

    // 4) final scalar
    k_final<<<1, 32, 0, stream>>>(ws, out);
}